// SupConLossNonDistributed_72559177498879
// MI455X (gfx1250) — compile-verified
//
#include <hip/hip_runtime.h>
#include <math.h>

typedef float v2f __attribute__((ext_vector_type(2)));
typedef float v8f __attribute__((ext_vector_type(8)));

#define N_TOTAL 8192
#define M_LAB   4096
#define DIMK    256
#define TEMP_INV 10.0f
#define ROWS    16
#define CHUNK   1024
#define LDS_STRIDE 1040              // row stride % 64 banks == 16 -> conflict-free halves
#define NCHUNK  (N_TOTAL / CHUNK)    // 8
#define TILES_PER_CHUNK (CHUNK / 16) // 64
#define WAVES   8
#define TILES_PER_WAVE (TILES_PER_CHUNK / WAVES) // 8
#define KSTEPS  (DIMK / 4)           // 64 WMMA f32 16x16x4 per tile

// ---------------- histogram of labels (1000 classes) ----------------
__global__ void supcon_hist_kernel(const int* __restrict__ labels, int* __restrict__ cnt) {
    const int t = threadIdx.x;           // blockDim.x == 1024, single block
    cnt[t] = 0;
    __syncthreads();
    for (int i = t; i < M_LAB; i += 1024)
        atomicAdd(&cnt[labels[i]], 1);
}

// ---------------- invrow[j] = 1 / (2*cnt(lab[j]) - 1) ----------------
__global__ void supcon_invrow_kernel(const int* __restrict__ labels,
                                     const int* __restrict__ cnt,
                                     float* __restrict__ invrow) {
    const int j = blockIdx.x * blockDim.x + threadIdx.x;
    if (j < M_LAB) {
        const int c = cnt[labels[j]];
        invrow[j] = 1.0f / (float)(2 * c - 1);
    }
}

// ---------------- fused GEMM (f32 WMMA) + online softmax reductions ----------------
__global__ __launch_bounds__(256) void supcon_main_kernel(
        const float* __restrict__ F,
        const int*   __restrict__ labels,
        const float* __restrict__ invrow,
        float*       __restrict__ rowloss) {
    __shared__ float S[ROWS * LDS_STRIDE];
    __shared__ int   rowlab[ROWS];

    const int t     = threadIdx.x;
    const int ibase = blockIdx.x * ROWS;
    const int lane  = t & 31;
    const int wave  = t >> 5;
    const int lmod  = lane & 15;   // M (A) / N (B) index within tile
    const int lhi   = lane >> 4;   // K half: +0 or +2

    if (t < ROWS) rowlab[t] = labels[(ibase + t) & (M_LAB - 1)];

    // Resident A fragments: 16x256 stripe, per-lane float2 per K-step of 4.
    // ISA 32-bit A 16x4 layout: lanes 0-15 -> K=0,1 ; lanes 16-31 -> K=2,3 ; M = lane&15.
    const float* aptr = F + (size_t)(ibase + lmod) * DIMK + 2 * lhi;
    v2f afrag[KSTEPS];
#pragma unroll
    for (int kk = 0; kk < KSTEPS; ++kk)
        afrag[kk] = *(const v2f*)(aptr + 4 * kk);

    // Reduction mapping: 16 threads per row.
    const int row = t >> 4;        // 0..15
    const int sub = t & 15;
    const int ig  = ibase + row;   // global row index
    float m = -INFINITY, s = 0.0f, Aacc = 0.0f, Bacc = 0.0f;

    __syncthreads();
    const int labi = rowlab[row];

    for (int ch = 0; ch < NCHUNK; ++ch) {
        const int jbase = ch * CHUNK;

        // ---- GEMM phase: each wave computes 8 16x16 tiles of this chunk ----
#pragma unroll 1
        for (int q = 0; q < TILES_PER_WAVE; ++q) {
            const int jt = wave * TILES_PER_WAVE + q;
            const int jg = jbase + jt * 16 + lmod;      // B column = global row jg of F
            const float* bptr = F + (size_t)jg * DIMK + 2 * lhi; // B 4x16: B[k,n]=F[n,k]
            v8f acc = {};
#pragma unroll
            for (int kk = 0; kk < KSTEPS; ++kk) {
                v2f b = *(const v2f*)(bptr + 4 * kk);
                acc = __builtin_amdgcn_wmma_f32_16x16x4_f32(
                        false, afrag[kk], false, b, (short)0, acc, false, false);
            }
            // C/D layout: VGPR r -> (M = r + 8*lhi, N = lmod)
            const int col = jt * 16 + lmod;
#pragma unroll
            for (int r = 0; r < 8; ++r)
                S[(r + 8 * lhi) * LDS_STRIDE + col] = acc[r];
        }
        __syncthreads();

        // ---- Reduction phase: online logsumexp + masked sums ----
        for (int step = 0; step < CHUNK / 16; ++step) {
            const int col = sub + (step << 4);
            const int j   = jbase + col;
            if (j == ig) continue;   // diagonal: mask=0 and exp(-1e9-m)==0 exactly
            const float l = S[row * LDS_STRIDE + col] * TEMP_INV;
            const float nm = fmaxf(m, l);
            s = s * expf(m - nm) + expf(l - nm);
            m = nm;
            const int jm = j & (M_LAB - 1);
            if (labels[jm] == labi) {
                const float ir = invrow[jm];   // divide by COLUMN j's rowsum (torch quirk)
                Aacc = fmaf(l, ir, Aacc);
                Bacc += ir;
            }
        }
        __syncthreads();
    }

    // ---- merge 16 partial states per row within 16-lane groups (wave32) ----
#pragma unroll
    for (int off = 8; off >= 1; off >>= 1) {
        const float om = __shfl_xor(m, off);
        const float os = __shfl_xor(s, off);
        const float oA = __shfl_xor(Aacc, off);
        const float oB = __shfl_xor(Bacc, off);
        const float nm = fmaxf(m, om);
        s = s * expf(m - nm) + os * expf(om - nm);
        m = nm;
        Aacc += oA;
        Bacc += oB;
    }
    if (sub == 0) {
        const float lse = m + logf(s);
        rowloss[ig] = Aacc - lse * Bacc;   // sum_j soft_label * log_proba for row ig
    }
}

// ---------------- deterministic final reduction ----------------
__global__ void supcon_finalize_kernel(const float* __restrict__ rowloss,
                                       float* __restrict__ out) {
    __shared__ float red[256];
    const int t = threadIdx.x;
    float p = 0.0f;
    for (int i = t; i < N_TOTAL; i += 256) p += rowloss[i];
    red[t] = p;
    __syncthreads();
    for (int w = 128; w > 0; w >>= 1) {
        if (t < w) red[t] += red[t + w];
        __syncthreads();
    }
    if (t == 0) out[0] = -(red[0] / (float)N_TOTAL) * 0.1f;
}

extern "C" void kernel_launch(void* const* d_in, const int* in_sizes, int n_in,
                              void* d_out, int out_size, void* d_ws, size_t ws_size,
                              hipStream_t stream) {
    const float* F      = (const float*)d_in[0];   // (8192, 256) f32
    const int*   labels = (const int*)d_in[1];     // (4096,) int32
    (void)in_sizes; (void)n_in; (void)out_size; (void)ws_size;

    char* ws = (char*)d_ws;
    int*   cnt     = (int*)(ws);                   // 1024 ints
    float* invrow  = (float*)(ws + 4096);          // 4096 floats
    float* rowloss = (float*)(ws + 4096 + 16384);  // 8192 floats

    supcon_hist_kernel<<<1, 1024, 0, stream>>>(labels, cnt);
    supcon_invrow_kernel<<<M_LAB / 256, 256, 0, stream>>>(labels, cnt, invrow);
    supcon_main_kernel<<<N_TOTAL / ROWS, 256, 0, stream>>>(F, labels, invrow, rowloss);
    supcon_finalize_kernel<<<1, 256, 0, stream>>>(rowloss, (float*)d_out);
}